// GIN_6356551598797
// MI455X (gfx1250) — compile-verified
//
#include <hip/hip_runtime.h>

#define N_NODES 50000
#define N_EDGES 800000
#define NF 128
#define BN_EPS 1e-5f
#define LDSTR 132   // padded LDS row stride (floats): 132 % 64 == 4 -> conflict-free lane fetches

typedef float v2f __attribute__((ext_vector_type(2)));
typedef float v8f __attribute__((ext_vector_type(8)));

// ---------------- K1: agg = x (copy), zero BN stats ----------------
__global__ void k_init(const float4* __restrict__ x4, float4* __restrict__ out4,
                       float* __restrict__ stats) {
  const int n4 = N_NODES * NF / 4;
  for (int i = blockIdx.x * blockDim.x + threadIdx.x; i < n4; i += gridDim.x * blockDim.x)
    out4[i] = x4[i];
  if (blockIdx.x == 0 && threadIdx.x < 2 * NF) stats[threadIdx.x] = 0.f;
}

// ---------------- K2: agg[dst] += x[src] over all edges ----------------
// One wave (32 lanes) per edge; each lane moves 4 contiguous floats (float4 gather,
// 4x global_atomic_add_f32 scatter). x and agg are L2-resident.
__global__ void k_scatter(const float4* __restrict__ x4, const long long* __restrict__ ei,
                          float* __restrict__ agg) {
  long long gid = (long long)blockIdx.x * blockDim.x + threadIdx.x;
  int e    = (int)(gid >> 5);
  int lane = (int)(gid & 31);
  if (e >= N_EDGES) return;
  int src = (int)ei[e];
  int dst = (int)ei[N_EDGES + e];
  float4 v = x4[src * (NF / 4) + lane];
  float* p = agg + (size_t)dst * NF + lane * 4;
  atomicAdd(p + 0, v.x);
  atomicAdd(p + 1, v.y);
  atomicAdd(p + 2, v.z);
  atomicAdd(p + 3, v.w);
}

// ---------------- shared WMMA inner product: 16x16 tile, K=128 via 32x f32 WMMA ----------------
// A (16x128) comes from padded LDS; B[k][n] = W[n][k] (torch [out,in] layout) read as
// contiguous float2 per lane from the weight row in global (cache-resident, 64KB).
__device__ __forceinline__ v8f wmma_tile_f32(const float* As, const float* __restrict__ W,
                                             int n0, int lane) {
  const int half = lane >> 4;   // 0: K pair {kk,kk+1}; 1: K pair {kk+2,kk+3}
  const int lm   = lane & 15;   // M for A, N for B
  const float* arow = As + lm * LDSTR + 2 * half;
  const float* wrow = W + (size_t)(n0 + lm) * NF + 2 * half;
  v8f acc = {0.f, 0.f, 0.f, 0.f, 0.f, 0.f, 0.f, 0.f};
#pragma unroll
  for (int kk = 0; kk < NF; kk += 4) {
    v2f a;
    a.x = arow[kk];
    a.y = arow[kk + 1];
    v2f b = *(const v2f*)(wrow + kk);
    acc = __builtin_amdgcn_wmma_f32_16x16x4_f32(false, a, false, b, (short)0, acc,
                                                false, false);
  }
  return acc;
}

// ---------------- K3: h = relu(agg @ W1^T + b1), in place on d_out ----------------
__global__ void k_gemm1(float* __restrict__ h, const float* __restrict__ W1,
                        const float* __restrict__ b1) {
  __shared__ float As[16 * LDSTR];
  const int m0 = blockIdx.x * 16;
  const int t  = threadIdx.x;
  for (int idx = t; idx < 16 * NF; idx += 256) {
    int m = idx >> 7, k = idx & 127;
    As[m * LDSTR + k] = h[(size_t)(m0 + m) * NF + k];
  }
  __syncthreads();
  const int wave = t >> 5, lane = t & 31;
  const int n0 = wave * 16;
  v8f acc = wmma_tile_f32(As, W1, n0, lane);
  const int half = lane >> 4, lm = lane & 15;
  const float bias = b1[n0 + lm];
#pragma unroll
  for (int r = 0; r < 8; ++r) {
    float v = acc[r] + bias;
    v = v > 0.f ? v : 0.f;
    h[(size_t)(m0 + r + 8 * half) * NF + n0 + lm] = v;
  }
}

// ---------------- K4: per-feature sum / sumsq of h ----------------
__global__ void k_stats(const float* __restrict__ h, float* __restrict__ stats) {
  __shared__ float red[512];
  const int f   = threadIdx.x & 127;
  const int sub = threadIdx.x >> 7;  // 0..1
  const int row0 = blockIdx.x * 128;
  float s = 0.f, q = 0.f;
  for (int i = sub; i < 128; i += 2) {
    int row = row0 + i;
    if (row < N_NODES) {
      float v = h[(size_t)row * NF + f];
      s += v;
      q += v * v;
    }
  }
  red[threadIdx.x]       = s;
  red[256 + threadIdx.x] = q;
  __syncthreads();
  if (threadIdx.x < 128) {
    atomicAdd(&stats[f],       red[f]       + red[128 + f]);
    atomicAdd(&stats[128 + f], red[256 + f] + red[384 + f]);
  }
}

// ---------------- K5: out = BN(h) @ W2^T + b2, in place on d_out ----------------
__global__ void k_bn_gemm2(float* __restrict__ h, const float* __restrict__ stats,
                           const float* __restrict__ gamma, const float* __restrict__ beta,
                           const float* __restrict__ W2, const float* __restrict__ b2) {
  __shared__ float As[16 * LDSTR];
  __shared__ float sc[128], sh[128];
  const int t = threadIdx.x;
  if (t < 128) {
    float mean = stats[t] * (1.f / N_NODES);
    float var  = stats[128 + t] * (1.f / N_NODES) - mean * mean;
    float inv  = rsqrtf(var + BN_EPS);
    float s    = gamma[t] * inv;
    sc[t] = s;
    sh[t] = beta[t] - mean * s;
  }
  __syncthreads();
  const int m0 = blockIdx.x * 16;
  for (int idx = t; idx < 16 * NF; idx += 256) {
    int m = idx >> 7, k = idx & 127;
    As[m * LDSTR + k] = h[(size_t)(m0 + m) * NF + k] * sc[k] + sh[k];
  }
  __syncthreads();
  const int wave = t >> 5, lane = t & 31;
  const int n0 = wave * 16;
  v8f acc = wmma_tile_f32(As, W2, n0, lane);
  const int half = lane >> 4, lm = lane & 15;
  const float bias = b2[n0 + lm];
#pragma unroll
  for (int r = 0; r < 8; ++r)
    h[(size_t)(m0 + r + 8 * half) * NF + n0 + lm] = acc[r] + bias;
}

extern "C" void kernel_launch(void* const* d_in, const int* in_sizes, int n_in,
                              void* d_out, int out_size, void* d_ws, size_t ws_size,
                              hipStream_t stream) {
  (void)in_sizes; (void)n_in; (void)out_size; (void)ws_size;
  const float*     x     = (const float*)d_in[0];
  const long long* ei    = (const long long*)d_in[1];
  const float*     W1    = (const float*)d_in[2];
  const float*     b1    = (const float*)d_in[3];
  const float*     gamma = (const float*)d_in[4];
  const float*     beta  = (const float*)d_in[5];
  const float*     W2    = (const float*)d_in[6];
  const float*     b2    = (const float*)d_in[7];
  float* out   = (float*)d_out;
  float* stats = (float*)d_ws;  // 256 floats: [0..127]=sum, [128..255]=sumsq

  k_init<<<2048, 256, 0, stream>>>((const float4*)x, (float4*)out, stats);

  long long tot = (long long)N_EDGES * 32;
  int sblocks = (int)((tot + 255) / 256);
  k_scatter<<<sblocks, 256, 0, stream>>>((const float4*)x, ei, out);

  k_gemm1<<<N_NODES / 16, 256, 0, stream>>>(out, W1, b1);

  k_stats<<<(N_NODES + 127) / 128, 256, 0, stream>>>(out, stats);

  k_bn_gemm2<<<N_NODES / 16, 256, 0, stream>>>(out, stats, gamma, beta, W2, b2);
}